// RegionalAttentionMapGenerator_15590731285133
// MI455X (gfx1250) — compile-verified
//
#include <hip/hip_runtime.h>

#define NB 64
#define NH 480
#define NW 864
#define ROWS_PER_BLK 16
#define CHUNKS (NH / ROWS_PER_BLK)   // 30
#define VEC_PER_ROW (NW / 4)         // 216
#define TILES_W (NW / 16)            // 54
#define TILES_H (NH / 16)            // 30

typedef __attribute__((ext_vector_type(2))) float v2f;
typedef __attribute__((ext_vector_type(8))) float v8f;

// ---------------- ws init (deterministic per launch) ----------------
__global__ void raag_init_ws(int* __restrict__ ws) {
    int b = threadIdx.x;
    if (b < NB) {
        int* s = ws + b * 8;
        s[0] = 0;      // cnt
        s[1] = NH;     // min_r identity (matches jnp.where(rows, r, H))
        s[2] = -1;     // max_r identity
        s[3] = NW;     // min_c identity
        s[4] = -1;     // max_c identity
    }
}

// ---------------- pass 1: streaming reduction (bandwidth bound) ----------------
__global__ void __launch_bounds__(256) raag_reduce(const float* __restrict__ mask,
                                                   int* __restrict__ ws) {
    __shared__ int s_cnt, s_minr, s_maxr, s_minc, s_maxc;
    const int b     = blockIdx.x / CHUNKS;
    const int chunk = blockIdx.x % CHUNKS;
    const int row0  = chunk * ROWS_PER_BLK;

    const float4* src =
        reinterpret_cast<const float4*>(mask + ((size_t)b * NH + row0) * NW);

    int cnt = 0, minr = NH, maxr = -1, minc = NW, maxc = -1;
    const int total = ROWS_PER_BLK * VEC_PER_ROW;   // 3456 float4 per block
    for (int i = threadIdx.x; i < total; i += 256) {
        const float4 v = src[i];
        const int r  = i / VEC_PER_ROW;
        const int c0 = (i - r * VEC_PER_ROW) * 4;
        const bool h0 = v.x > 0.5f, h1 = v.y > 0.5f, h2 = v.z > 0.5f, h3 = v.w > 0.5f;
        const int hc = (int)h0 + (int)h1 + (int)h2 + (int)h3;
        cnt += hc;
        if (hc) {
            const int rr = row0 + r;
            minr = min(minr, rr); maxr = max(maxr, rr);
            if (h0) { minc = min(minc, c0 + 0); maxc = max(maxc, c0 + 0); }
            if (h1) { minc = min(minc, c0 + 1); maxc = max(maxc, c0 + 1); }
            if (h2) { minc = min(minc, c0 + 2); maxc = max(maxc, c0 + 2); }
            if (h3) { minc = min(minc, c0 + 3); maxc = max(maxc, c0 + 3); }
        }
    }

    if (threadIdx.x == 0) { s_cnt = 0; s_minr = NH; s_maxr = -1; s_minc = NW; s_maxc = -1; }
    __syncthreads();
    if (cnt) {
        atomicAdd(&s_cnt, cnt);
        atomicMin(&s_minr, minr); atomicMax(&s_maxr, maxr);
        atomicMin(&s_minc, minc); atomicMax(&s_maxc, maxc);
    }
    __syncthreads();
    if (threadIdx.x == 0 && s_cnt) {
        int* g = ws + b * 8;
        atomicAdd(&g[0], s_cnt);
        atomicMin(&g[1], s_minr); atomicMax(&g[2], s_maxr);
        atomicMin(&g[3], s_minc); atomicMax(&g[4], s_maxc);
    }
}

// ---------------- pass 2: finalize bbox (tiny) ----------------
__global__ void raag_finalize(const int* __restrict__ ws,
                              int* __restrict__ bbox_ws,
                              int* __restrict__ bbox_out,
                              const int* __restrict__ n_pts,
                              const int* __restrict__ n_loose) {
    const int b = threadIdx.x;
    if (b >= NB) return;
    const int* s = ws + b * 8;
    const int cnt = s[0], minr = s[1], maxr = s[2], minc = s[3], maxc = s[4];
    const int loose = *n_loose;
    const bool full = cnt < *n_pts;
    const int y0 = full ? 0        : min(max(minr - loose, 0), NH - 1);
    const int y1 = full ? (NH - 1) : min(max(maxr + loose, 0), NH - 1);
    const int x0 = full ? 0        : min(max(minc - loose, 0), NW - 1);
    const int x1 = full ? (NW - 1) : min(max(maxc + loose, 0), NW - 1);
    bbox_ws[b * 4 + 0] = y0; bbox_ws[b * 4 + 1] = y1;
    bbox_ws[b * 4 + 2] = x0; bbox_ws[b * 4 + 3] = x1;
    bbox_out[b * 4 + 0] = y0; bbox_out[b * 4 + 1] = y1;
    bbox_out[b * 4 + 2] = x0; bbox_out[b * 4 + 3] = x1;
}

// ---------------- pass 3: WMMA outer-product writer ----------------
// att_map[b] = rowflags ⊗ colflags. One V_WMMA_F32_16X16X4_F32 per 16x16 tile:
// A(16x4) = [rowflags,0,0,0], B(4x16) = [colflags;0;0;0]; only K=0 is nonzero,
// which per ISA layout lives in VGPR0 lanes 0-15 for both A and B.
__global__ void __launch_bounds__(256) raag_map(float* __restrict__ out,
                                                const int* __restrict__ bbox_ws) {
    const int b  = blockIdx.x / TILES_H;
    const int tr = blockIdx.x % TILES_H;
    const int r0 = tr * 16;
    const int y0 = bbox_ws[b * 4 + 0], y1 = bbox_ws[b * 4 + 1];
    const int x0 = bbox_ws[b * 4 + 2], x1 = bbox_ws[b * 4 + 3];

    const int lane = threadIdx.x & 31;
    const int wave = threadIdx.x >> 5;
    const int l16  = lane & 15;

    const int r = r0 + l16;
    const float rflag = ((lane < 16) && (r >= y0) && (r <= y1)) ? 1.0f : 0.0f;
    v2f a; a.x = rflag; a.y = 0.0f;

    float* samp = out + (size_t)b * NH * NW;
    const int ro = r0 + ((lane >> 4) << 3);   // store row base for this lane-half

    for (int tc = wave; tc < TILES_W; tc += 8) {   // wave-uniform bound: EXEC stays all-1s
        const int c0 = tc * 16;
        const int c  = c0 + l16;
        const float cflag = ((lane < 16) && (c >= x0) && (c <= x1)) ? 1.0f : 0.0f;
        v2f bv; bv.x = cflag; bv.y = 0.0f;

        v8f acc = {};
        acc = __builtin_amdgcn_wmma_f32_16x16x4_f32(
            /*neg_a=*/false, a, /*neg_b=*/false, bv,
            /*c_mod=*/(short)0, acc, /*reuse_a=*/false, /*reuse_b=*/false);

        // C/D layout: VGPR j -> row M=j (lanes 0-15) or M=j+8 (lanes 16-31), N = lane%16
        float* dst = samp + (size_t)ro * NW + c0 + l16;
#pragma unroll
        for (int j = 0; j < 8; ++j)
            dst[(size_t)j * NW] = acc[j];
    }
}

extern "C" void kernel_launch(void* const* d_in, const int* in_sizes, int n_in,
                              void* d_out, int out_size, void* d_ws, size_t ws_size,
                              hipStream_t stream) {
    const float* mask    = (const float*)d_in[0];
    const int*   n_pts   = (const int*)d_in[1];
    const int*   n_loose = (const int*)d_in[2];

    float* outp     = (float*)d_out;
    int*   bbox_out = (int*)(outp + (size_t)NB * NH * NW);  // raw int32 tail of d_out

    int* ws      = (int*)d_ws;      // NB*8 ints: {cnt,minr,maxr,minc,maxc,pad..}
    int* bbox_ws = ws + NB * 8;     // NB*4 ints: finalized {y0,y1,x0,x1}

    raag_init_ws<<<1, 64, 0, stream>>>(ws);
    raag_reduce<<<NB * CHUNKS, 256, 0, stream>>>(mask, ws);
    raag_finalize<<<1, 64, 0, stream>>>(ws, bbox_ws, bbox_out, n_pts, n_loose);
    raag_map<<<NB * TILES_H, 256, 0, stream>>>(outp, bbox_ws);
}